// SimpleHeteroGNN_26663156973731
// MI455X (gfx1250) — compile-verified
//
#include <hip/hip_runtime.h>
#include <hip/hip_bf16.h>

typedef __attribute__((ext_vector_type(2))) float v2f;
typedef __attribute__((ext_vector_type(8))) float v8f;

#define D_IN   256
#define D_OUT  128
#define XPAD   260                    // padded row stride (dwords): 4-bank advance/row -> conflict-free b64
#define N_MOVIE 100000
#define N_DIR   50000
#define N_ACT   150000

// ---------------------------------------------------------------------------
// C[M,128] = X[M,256] @ W[256,128] (+ bias), fp32 WMMA 16x16x4.
// 8 waves/block; wave w owns the 16-wide column strip w*16.
// B panel (64 x v2f = 128 VGPRs) lives in registers for the whole block.
// X tiles double-buffered in LDS, staged with global_load_async_to_lds_b128.
// ---------------------------------------------------------------------------
__global__ __launch_bounds__(256) void gemm_bias_kernel(
    const float* __restrict__ X, const float* __restrict__ W,
    const float* __restrict__ bias, float* __restrict__ C, int M)
{
    __shared__ __align__(16) float lds_x[2][16 * XPAD];   // 2 x 16,640 B

    const int tid     = threadIdx.x;
    const int lane    = tid & 31;
    const int wave    = tid >> 5;          // 0..7 => column strip
    const int halfSel = lane >> 4;         // 0: K=0..1/M=0..7 ; 1: K=2..3/M=8..15
    const int l16     = lane & 15;
    const int col     = (wave << 4) + l16;

    // ---- B panel in registers: breg[i] = {W[4i+2h][col], W[4i+1+2h][col]} ----
    v2f breg[64];
    #pragma unroll
    for (int i = 0; i < 64; ++i) {
        const int k = (i << 2) + (halfSel << 1);
        breg[i].x = W[(size_t)k * D_OUT + col];
        breg[i].y = W[(size_t)(k + 1) * D_OUT + col];
    }

    const float bval = bias ? bias[col] : 0.0f;
    const int numTiles = (M + 15) >> 4;

    // issue async stage of a 16x256 tile into buffer `buf` (4 b128 ops/thread)
    auto stage = [&](int buf, int tile) {
        const int row0 = tile << 4;
        #pragma unroll
        for (int i = 0; i < 4; ++i) {
            const int idx = tid + (i << 8);        // 0..1023 float4 slots
            const int r   = idx >> 6;
            const int c4  = (idx & 63) << 2;
            int gr = row0 + r;
            if (gr >= M) gr = M - 1;               // clamp: junk rows are store-guarded
            const float* gp = X + (size_t)gr * D_IN + c4;
            const unsigned lo = (unsigned)(size_t)(const void*)&lds_x[buf][r * XPAD + c4];
            asm volatile("global_load_async_to_lds_b128 %0, %1, off"
                         :: "v"(lo), "v"(gp) : "memory");
        }
    };

    int t = blockIdx.x;
    if (t >= numTiles) return;                     // uniform per block
    int cur = 0;
    stage(cur, t);                                 // prologue: tile t in flight

    for (; t < numTiles; t += gridDim.x) {
        // stage next tile into the other buffer (clamped so count stays uniform)
        const int nxt = (t + (int)gridDim.x < numTiles) ? t + (int)gridDim.x : t;
        stage(cur ^ 1, nxt);
        // retire the oldest batch (current tile), keep next batch in flight
        asm volatile("s_wait_asynccnt 0x4" ::: "memory");
        __syncthreads();

        v8f acc = {0.f, 0.f, 0.f, 0.f, 0.f, 0.f, 0.f, 0.f};
        // A frag: lane<16 holds X[l16][k..k+1]; lane>=16 holds X[l16][k+2..k+3]
        const float* ax = &lds_x[cur][l16 * XPAD + (halfSel << 1)];
        #pragma unroll
        for (int i = 0; i < 64; ++i) {
            const v2f a = *(const v2f*)(ax + (i << 2));   // ds_load_b64, conflict-free
            acc = __builtin_amdgcn_wmma_f32_16x16x4_f32(
                false, a, false, breg[i], (short)0, acc, false, false);
        }

        // C/D layout: acc[j] = tile row (j + 8*halfSel), col l16
        const int row0 = t << 4;
        #pragma unroll
        for (int j = 0; j < 8; ++j) {
            const int gr = row0 + (halfSel << 3) + j;
            if (gr < M) C[(size_t)gr * D_OUT + col] = acc[j] + bval;
        }
        __syncthreads();                           // all reads of buf done before reuse
        cur ^= 1;
    }
}

// ---------------------------------------------------------------------------
// W_comb = Wa + Wb ; b_comb = ba + bb   (movie has two root terms)
// ---------------------------------------------------------------------------
__global__ void combine_root_kernel(const float* __restrict__ Wa,
                                    const float* __restrict__ Wb,
                                    const float* __restrict__ ba,
                                    const float* __restrict__ bb,
                                    float* __restrict__ Wc,
                                    float* __restrict__ bc)
{
    const int i = blockIdx.x * blockDim.x + threadIdx.x;
    if (i < D_IN * D_OUT) Wc[i] = Wa[i] + Wb[i];
    if (i < D_OUT)        bc[i] = ba[i] + bb[i];
}

// ---------------------------------------------------------------------------
// Edge scatter: one wave per edge; lane L handles floats [4L,4L+4).
// out[dst] += y[src]  via native fp32 global atomics (no CAS loop).
// ---------------------------------------------------------------------------
__global__ __launch_bounds__(256) void scatter_edges_kernel(
    const float* __restrict__ y, const long long* __restrict__ edges,
    int E, float* __restrict__ out)
{
    const long long gid  = (long long)blockIdx.x * blockDim.x + threadIdx.x;
    const int       lane = (int)(gid & 31);
    const long long e    = gid >> 5;
    if (e >= E) return;

    const long long s = edges[e];        // src node id
    const long long d = edges[E + e];    // dst node id

    const float4 v = *(const float4*)(y + (size_t)s * D_OUT + (lane << 2));
    float* p = out + (size_t)d * D_OUT + (lane << 2);

    asm volatile("global_atomic_add_f32 %0, %1, off" :: "v"(p + 0), "v"(v.x) : "memory");
    asm volatile("global_atomic_add_f32 %0, %1, off" :: "v"(p + 1), "v"(v.y) : "memory");
    asm volatile("global_atomic_add_f32 %0, %1, off" :: "v"(p + 2), "v"(v.z) : "memory");
    asm volatile("global_atomic_add_f32 %0, %1, off" :: "v"(p + 3), "v"(v.w) : "memory");
}

// ---------------------------------------------------------------------------
// In-place ReLU over the whole concatenated output (float4 vectorized).
// ---------------------------------------------------------------------------
__global__ __launch_bounds__(256) void relu_kernel(float* __restrict__ o, long long n4)
{
    const long long i = (long long)blockIdx.x * blockDim.x + threadIdx.x;
    if (i >= n4) return;
    float4 v = *(float4*)(o + (i << 2));
    v.x = fmaxf(v.x, 0.f); v.y = fmaxf(v.y, 0.f);
    v.z = fmaxf(v.z, 0.f); v.w = fmaxf(v.w, 0.f);
    *(float4*)(o + (i << 2)) = v;
}

// ---------------------------------------------------------------------------
extern "C" void kernel_launch(void* const* d_in, const int* in_sizes, int n_in,
                              void* d_out, int out_size, void* d_ws, size_t ws_size,
                              hipStream_t stream)
{
    const float* x_movie = (const float*)d_in[0];
    const float* x_dir   = (const float*)d_in[1];
    const float* x_act   = (const float*)d_in[2];
    const long long* e_md = (const long long*)d_in[3];
    const long long* e_dm = (const long long*)d_in[4];
    const long long* e_ma = (const long long*)d_in[5];
    const long long* e_am = (const long long*)d_in[6];
    const float* W_rel_md  = (const float*)d_in[7];
    const float* b_md      = (const float*)d_in[8];
    const float* W_root_md = (const float*)d_in[9];
    const float* W_rel_dm  = (const float*)d_in[10];
    const float* b_dm      = (const float*)d_in[11];
    const float* W_root_dm = (const float*)d_in[12];
    const float* W_rel_ma  = (const float*)d_in[13];
    const float* b_ma      = (const float*)d_in[14];
    const float* W_root_ma = (const float*)d_in[15];
    const float* W_rel_am  = (const float*)d_in[16];
    const float* b_am      = (const float*)d_in[17];
    const float* W_root_am = (const float*)d_in[18];

    float* out_movie = (float*)d_out;
    float* out_dir   = out_movie + (size_t)N_MOVIE * D_OUT;
    float* out_act   = out_dir   + (size_t)N_DIR   * D_OUT;

    // workspace: one reusable projection buffer (max src = 150k rows) + W_comb/b_comb
    float* y      = (float*)d_ws;
    float* W_comb = y + (size_t)N_ACT * D_OUT;
    float* b_comb = W_comb + D_IN * D_OUT;

    const int E = in_sizes[3] / 2;   // 300000

    auto gemm = [&](const float* X, const float* W, const float* b, float* C, int M) {
        const int tiles = (M + 15) >> 4;
        const int grid  = tiles < 2048 ? tiles : 2048;
        gemm_bias_kernel<<<grid, 256, 0, stream>>>(X, W, b, C, M);
    };
    auto scatter = [&](const float* yy, const long long* e, float* o) {
        const long long thr = (long long)E * 32;
        scatter_edges_kernel<<<(int)((thr + 255) / 256), 256, 0, stream>>>(yy, e, E, o);
    };

    // 1. combined movie root weights
    combine_root_kernel<<<(D_IN * D_OUT + 255) / 256, 256, 0, stream>>>(
        W_root_dm, W_root_am, b_dm, b_am, W_comb, b_comb);

    // 2. root GEMMs initialize the outputs (full overwrite incl. bias)
    gemm(x_movie, W_comb,    b_comb, out_movie, N_MOVIE);
    gemm(x_dir,   W_root_md, b_md,   out_dir,   N_DIR);
    gemm(x_act,   W_root_ma, b_ma,   out_act,   N_ACT);

    // 3. per relation: project source features (256->128), then atomic scatter
    gemm(x_dir,   W_rel_dm, nullptr, y, N_DIR);    scatter(y, e_dm, out_movie);
    gemm(x_act,   W_rel_am, nullptr, y, N_ACT);    scatter(y, e_am, out_movie);
    gemm(x_movie, W_rel_md, nullptr, y, N_MOVIE);  scatter(y, e_md, out_dir);
    gemm(x_movie, W_rel_ma, nullptr, y, N_MOVIE);  scatter(y, e_ma, out_act);

    // 4. ReLU over the whole concatenated output
    const long long n4 = (long long)out_size / 4;
    relu_kernel<<<(int)((n4 + 255) / 256), 256, 0, stream>>>((float*)d_out, n4);
}